// CoordinateDescentRouter_67310727463356
// MI455X (gfx1250) — compile-verified
//
#include <hip/hip_runtime.h>
#include <hip/hip_bf16.h>
#include <math.h>

// ---------------------------------------------------------------------------
// Problem constants (match reference setup_inputs):
//   x: [B=8, N=8192, D=2048] f32, routing_token: [1, 2048] f32, num_tokens=1024
//   effective_k = min(1024*9/8, 8192) = 1152, eps = 1, iters = 50
// ---------------------------------------------------------------------------
constexpr int B_DIM   = 8;
constexpr int N_DIM   = 8192;
constexpr int D_DIM   = 2048;
constexpr int KOUT    = 1024;
constexpr int N_ITERS = 50;

typedef __attribute__((ext_vector_type(2))) float v2f;
typedef __attribute__((ext_vector_type(4))) float v4f;
typedef __attribute__((ext_vector_type(8))) float v8f;

// ---------------------------------------------------------------------------
// Phase 1: s[b,n] = dot(x[b,n,:], r[:])  via V_WMMA_F32_16X16X4_F32.
// One wave computes 16 consecutive rows; lane = m + 16*h.
//
// K-assignment trick: the WMMA's 4 K-slots (vgpr0/1 x lane-half h) can hold
// ANY K-values as long as A and B agree per slot (dot is commutative over K).
// So each lane does ONE b128 load of x[m, k0+4h .. k0+4h+3] and feeds:
//   WMMA#1 slots: vgpr0 <- av.x (K=k0+4h),   vgpr1 <- av.y (K=k0+4h+1)
//   WMMA#2 slots: vgpr0 <- av.z (K=k0+4h+2), vgpr1 <- av.w (K=k0+4h+3)
// B carries the routing token only in column N=0 (lanes 0 and 16), fetched
// from LDS at the matching K positions.
// D-tile: column N=0 -> lane 0 holds M=0..7 in c[0..7], lane 16 holds M=8..15.
// ---------------------------------------------------------------------------
__global__ void __launch_bounds__(256)
router_gemv_wmma(const float* __restrict__ x, const float* __restrict__ r,
                 float* __restrict__ s_out) {
    __shared__ float r_sh[D_DIM];
    for (int i = threadIdx.x; i < D_DIM; i += blockDim.x) r_sh[i] = r[i];
    __syncthreads();

    const int lane = threadIdx.x & 31;
    const int wave = blockIdx.x * (blockDim.x >> 5) + (threadIdx.x >> 5);
    const int m    = lane & 15;      // row within tile / N-col for B
    const int h    = lane >> 4;      // lane-half (K-slot group)
    const bool n0  = (m == 0);       // lane holds B column 0

    const size_t row  = (size_t)wave * 16 + (size_t)m;
    const float* xrow = x + row * (size_t)D_DIM + 4 * h;   // 16B-aligned stream
    const float* rrow = r_sh + 4 * h;

    v8f c = {};
#pragma unroll 4
    for (int k0 = 0; k0 < D_DIM; k0 += 8) {
        if ((k0 & 31) == 0)
            __builtin_prefetch(xrow + k0 + 512, 0, 1);     // ~2KB ahead
        v4f av = *(const v4f*)(xrow + k0);                 // one b128 per lane
        v4f rv = *(const v4f*)(rrow + k0);                 // matching K from LDS

        v2f a01 = {av.x, av.y};
        v2f b01 = {n0 ? rv.x : 0.0f, n0 ? rv.y : 0.0f};
        c = __builtin_amdgcn_wmma_f32_16x16x4_f32(
                false, a01, false, b01, (short)0, c, false, false);

        v2f a23 = {av.z, av.w};
        v2f b23 = {n0 ? rv.z : 0.0f, n0 ? rv.w : 0.0f};
        c = __builtin_amdgcn_wmma_f32_16x16x4_f32(
                false, a23, false, b23, (short)0, c, false, false);
    }

    if (m == 0) {  // lanes 0 and 16 hold the result column
        float* srow = s_out + (size_t)wave * 16 + (size_t)h * 8;
#pragma unroll
        for (int j = 0; j < 8; ++j) srow[j] = c[j];
    }
}

// ---------------------------------------------------------------------------
// Block reductions (wave32 shuffles + 32-entry LDS stage, 1024 threads)
// ---------------------------------------------------------------------------
__device__ __forceinline__ float warp_max32(float v) {
#pragma unroll
    for (int o = 16; o > 0; o >>= 1) v = fmaxf(v, __shfl_xor(v, o, 32));
    return v;
}
__device__ __forceinline__ float warp_sum32(float v) {
#pragma unroll
    for (int o = 16; o > 0; o >>= 1) v += __shfl_xor(v, o, 32);
    return v;
}

__device__ float block_max(float v, float* red) {
    const int wid = threadIdx.x >> 5, lane = threadIdx.x & 31;
    v = warp_max32(v);
    if (lane == 0) red[wid] = v;
    __syncthreads();
    if (wid == 0) {
        float t = red[lane];           // exactly 32 waves of 32 lanes
        t = warp_max32(t);
        if (lane == 0) red[0] = t;
    }
    __syncthreads();
    float res = red[0];
    __syncthreads();
    return res;
}

__device__ float block_sum(float v, float* red) {
    const int wid = threadIdx.x >> 5, lane = threadIdx.x & 31;
    v = warp_sum32(v);
    if (lane == 0) red[wid] = v;
    __syncthreads();
    if (wid == 0) {
        float t = red[lane];
        t = warp_sum32(t);
        if (lane == 0) red[0] = t;
    }
    __syncthreads();
    float res = red[0];
    __syncthreads();
    return res;
}

// ---------------------------------------------------------------------------
// Phase 2+3: coordinate descent (only scalar `a` needs carrying since
// s + b_prev = min(s, -a_prev)) then exact top-1024 by bitonic sort of
// 64-bit keys in LDS (desc score, asc index tie-break, matching lax.top_k).
// Straight-through trick collapses forward values to exactly 1.0.
// One block per batch row: 1024 threads x 8 elements; 64KB LDS keys.
// ---------------------------------------------------------------------------
__global__ void __launch_bounds__(1024)
router_descent_topk(const float* __restrict__ s_in,
                    float* __restrict__ out_vals, int* __restrict__ out_idx) {
    constexpr int PER = N_DIM / 1024;  // 8 elements per thread
    extern __shared__ unsigned long long keys[];  // 8192 * 8B dynamic LDS
    __shared__ float red[32];

    const int tid = threadIdx.x;
    const float* srow = s_in + (size_t)blockIdx.x * N_DIM;

    float sv[PER];
#pragma unroll
    for (int j = 0; j < PER; ++j) sv[j] = srow[j * 1024 + tid];

    const float logk = logf(1152.0f);          // log(effective_k)
    // Iteration 1 in closed form: b0 = -s -> sb = 0 -> lse = log(N)
    float a = logk - logf((float)N_DIM);

    for (int it = 1; it < N_ITERS; ++it) {
        const float na = -a;
        float lm = -3.402823466e38f;
#pragma unroll
        for (int j = 0; j < PER; ++j) lm = fmaxf(lm, fminf(sv[j], na));
        const float mx = block_max(lm, red);
        float ls = 0.0f;
#pragma unroll
        for (int j = 0; j < PER; ++j) ls += expf(fminf(sv[j], na) - mx);
        const float tot = block_sum(ls, red);
        a = logk - (mx + logf(tot));
    }

    // scores = exp(min(s + a, 0)) in (0,1]; build sortable composite keys
#pragma unroll
    for (int j = 0; j < PER; ++j) {
        const int i = j * 1024 + tid;
        const float sc = expf(fminf(sv[j] + a, 0.0f));
        const unsigned ub = __float_as_uint(sc);   // positive float: monotonic
        keys[i] = (((unsigned long long)(0x7FFFFFFFu - ub)) << 32) |
                  (unsigned long long)(unsigned)i;
    }
    __syncthreads();

    // Ascending bitonic sort of 8192 keys (=> descending score, asc index)
    for (unsigned k = 2; k <= (unsigned)N_DIM; k <<= 1) {
        for (unsigned j = k >> 1; j > 0; j >>= 1) {
            for (unsigned i = tid; i < (unsigned)N_DIM; i += 1024) {
                const unsigned ixj = i ^ j;
                if (ixj > i) {
                    unsigned long long a0 = keys[i], a1 = keys[ixj];
                    const bool up = ((i & k) == 0);
                    if ((a0 > a1) == up) { keys[i] = a1; keys[ixj] = a0; }
                }
            }
            __syncthreads();
        }
    }

    // First 1024 keys are the selection
    out_vals[blockIdx.x * KOUT + tid] = 1.0f;  // straight-through forward value
    out_idx [blockIdx.x * KOUT + tid] = (int)(keys[tid] & 0xFFFFFFFFull);
}

// ---------------------------------------------------------------------------
extern "C" void kernel_launch(void* const* d_in, const int* in_sizes, int n_in,
                              void* d_out, int out_size, void* d_ws, size_t ws_size,
                              hipStream_t stream) {
    const float* x = (const float*)d_in[0];        // [8, 8192, 2048] f32
    const float* r = (const float*)d_in[1];        // [1, 2048] f32
    (void)in_sizes; (void)n_in; (void)out_size; (void)ws_size;

    float* s_ws = (float*)d_ws;                    // [65536] f32 scores scratch

    // Phase 1: 65536 rows / 16 rows-per-wave / 8 waves-per-block = 512 blocks
    router_gemv_wmma<<<512, 256, 0, stream>>>(x, r, s_ws);

    // Phase 2+3: one block per batch row; 64KB dynamic LDS for sort keys
    float* out_vals = (float*)d_out;
    int*   out_idx  = (int*)(out_vals + (size_t)B_DIM * KOUT);
    const size_t lds_bytes = (size_t)N_DIM * sizeof(unsigned long long);
    router_descent_topk<<<B_DIM, 1024, lds_bytes, stream>>>(s_ws, out_vals, out_idx);
}